// Chowder_16080357556255
// MI455X (gfx1250) — compile-verified
//
#include <hip/hip_runtime.h>

typedef __attribute__((ext_vector_type(2))) float v2f;
typedef __attribute__((ext_vector_type(4))) float v4f;
typedef __attribute__((ext_vector_type(8))) float v8f;

#define BB 8
#define NN 50000
#define LL 512
#define RR 5
#define CC 2
#define H1 200
#define H2 100

// ---------------------------------------------------------------------------
// Kernel 1: scores[b,n] = dot(x[b,n,:], conv_w) + conv_b   (bandwidth bound)
// One wave32 per instance. Lane loads float4 (global_load_b128), 4 iters =
// 512 elems. Non-temporal: 819 MB streamed once, no reuse (L2 is 192 MB).
// Roofline: 819.2 MB / 23.3 TB/s ~= 35 us.
// ---------------------------------------------------------------------------
__global__ __launch_bounds__(256) void chowder_scores(
    const float* __restrict__ x, const float* __restrict__ conv_w,
    const float* __restrict__ conv_b, float* __restrict__ scores) {
  const int gwave = (blockIdx.x * blockDim.x + threadIdx.x) >> 5;
  const int lane  = threadIdx.x & 31;
  if (gwave >= BB * NN) return;

  const v4f* xp = (const v4f*)(x + (size_t)gwave * LL);
  const v4f* wp = (const v4f*)conv_w;

  float acc = 0.f;
#pragma unroll
  for (int it = 0; it < 4; ++it) {
    v4f xv = __builtin_nontemporal_load(&xp[it * 32 + lane]);
    v4f wv = wp[it * 32 + lane];  // 2 KB, cache resident
    acc += xv.x * wv.x + xv.y * wv.y + xv.z * wv.z + xv.w * wv.w;
  }
#pragma unroll
  for (int off = 16; off >= 1; off >>= 1) acc += __shfl_xor(acc, off, 32);

  if (lane == 0) scores[gwave] = acc + conv_b[0];
}

// ---------------------------------------------------------------------------
// Kernel 2: per batch row, top-5 (descending) and bottom-5 (ascending).
// Per-thread sorted insertion over strided chunk, then LDS tree merge of
// sorted 5-lists. feat layout per batch: [min0..min4, max0..max4].
// ---------------------------------------------------------------------------
__device__ inline void merge_desc(float* a, const float* b) {
  float o[RR];
  int i = 0, j = 0;
#pragma unroll
  for (int k = 0; k < RR; ++k)
    o[k] = (j >= RR || (i < RR && a[i] >= b[j])) ? a[i++] : b[j++];
#pragma unroll
  for (int k = 0; k < RR; ++k) a[k] = o[k];
}
__device__ inline void merge_asc(float* a, const float* b) {
  float o[RR];
  int i = 0, j = 0;
#pragma unroll
  for (int k = 0; k < RR; ++k)
    o[k] = (j >= RR || (i < RR && a[i] <= b[j])) ? a[i++] : b[j++];
#pragma unroll
  for (int k = 0; k < RR; ++k) a[k] = o[k];
}

__global__ __launch_bounds__(256) void chowder_topk(
    const float* __restrict__ scores, float* __restrict__ feat) {
  __shared__ float smax[256][RR];
  __shared__ float smin[256][RR];
  const int b = blockIdx.x;
  const int t = threadIdx.x;

  float mx[RR], mn[RR];
#pragma unroll
  for (int i = 0; i < RR; ++i) { mx[i] = -INFINITY; mn[i] = INFINITY; }

  for (int n = t; n < NN; n += 256) {
    float s = scores[b * NN + n];
    if (s > mx[RR - 1]) {
      mx[RR - 1] = s;
#pragma unroll
      for (int i = RR - 1; i > 0; --i)
        if (mx[i] > mx[i - 1]) { float tp = mx[i]; mx[i] = mx[i - 1]; mx[i - 1] = tp; }
    }
    if (s < mn[RR - 1]) {
      mn[RR - 1] = s;
#pragma unroll
      for (int i = RR - 1; i > 0; --i)
        if (mn[i] < mn[i - 1]) { float tp = mn[i]; mn[i] = mn[i - 1]; mn[i - 1] = tp; }
    }
  }
#pragma unroll
  for (int i = 0; i < RR; ++i) { smax[t][i] = mx[i]; smin[t][i] = mn[i]; }
  __syncthreads();

  for (int stride = 128; stride >= 1; stride >>= 1) {
    if (t < stride) {
      float a[RR], c[RR];
#pragma unroll
      for (int i = 0; i < RR; ++i) { a[i] = smax[t][i]; c[i] = smax[t + stride][i]; }
      merge_desc(a, c);
#pragma unroll
      for (int i = 0; i < RR; ++i) smax[t][i] = a[i];
#pragma unroll
      for (int i = 0; i < RR; ++i) { a[i] = smin[t][i]; c[i] = smin[t + stride][i]; }
      merge_asc(a, c);
#pragma unroll
      for (int i = 0; i < RR; ++i) smin[t][i] = a[i];
    }
    __syncthreads();
  }
  if (t == 0) {
#pragma unroll
    for (int i = 0; i < RR; ++i) {
      feat[b * 2 * RR + i]      = smin[0][i];  // ascending bottom-5
      feat[b * 2 * RR + RR + i] = smax[0][i];  // descending top-5
    }
  }
}

// ---------------------------------------------------------------------------
// Kernel 3: MLP [8,10] @ w1[10,200] + b1 -> @ w2[200,100] + b2 -> @ w3[100,2]
// Single wave32, V_WMMA_F32_16X16X4_F32. M padded 8->16, K stepped by 4,
// N tiled by 16. Weight tiles are staged zero-padded into LDS as interleaved
// K-pairs so each WMMA operand is one aligned ds_load_b64 — no conditionals
// (and no EXEC manipulation) inside the WMMA loops.
//
// LDS B-tile pair layout: wt[((k>>1)*16 + col)*2 + (k&1)], k even-aligned,
// so B operand {w[k][col], w[k+1][col]} is contiguous 8 bytes.
// ---------------------------------------------------------------------------
#define H1P 208   // 13 * 16
#define H2P 112   // 7 * 16

__device__ __forceinline__ void stage_btile(
    float* __restrict__ wt, const float* __restrict__ wsrc, int K, int ncols,
    int tile, int lane) {
  // wt gets K x 16 (K even) zero-padded tile of wsrc[K][ncols], cols from tile*16.
  const int total = K * 16;
  for (int i = lane; i < total; i += 32) {
    const int k = i >> 4, col = i & 15;
    const int gcol = tile * 16 + col;
    wt[(((k >> 1) * 16 + col) << 1) + (k & 1)] =
        (gcol < ncols) ? wsrc[k * ncols + gcol] : 0.f;
  }
}

__global__ __launch_bounds__(32) void chowder_mlp(
    const float* __restrict__ feat, const float* __restrict__ w1,
    const float* __restrict__ b1, const float* __restrict__ w2,
    const float* __restrict__ b2, const float* __restrict__ w3,
    const float* __restrict__ b3, float* __restrict__ out) {
  __shared__ float fpad[16][12];   // features, K padded 10->12, rows 8..15 zero
  __shared__ float h1[16][H1P];    // layer-1 out
  __shared__ float h2[16][H2P];    // layer-2 out
  __shared__ float wt[100 * 32];   // staged B tile (pairs), max 200x16 floats
  __shared__ float bt[16];         // staged bias tile

  const int lane = threadIdx.x;    // 0..31
  const int m    = lane & 15;      // A row within 16-row tile
  const int hi   = lane >> 4;      // 0: K pair {0,1}; 1: K pair {2,3}
  const int col0 = lane & 15;      // B/D column within 16-col tile

  // Zero LDS pads, load features (K columns 10,11 and rows 8..15 stay zero).
  for (int i = lane; i < 16 * 12; i += 32) ((float*)fpad)[i] = 0.f;
  for (int i = lane; i < 16 * H1P; i += 32) ((float*)h1)[i] = 0.f;
  for (int i = lane; i < 16 * H2P; i += 32) ((float*)h2)[i] = 0.f;
  __syncthreads();
  for (int i = lane; i < BB * 2 * RR; i += 32)
    fpad[i / (2 * RR)][i % (2 * RR)] = feat[i];
  __syncthreads();

  // ---- Layer 1: [16 x 12] @ [12 x 200] -> h1[16 x 208] ----
  for (int t = 0; t < 13; ++t) {
    // Stage 12x16 tile of w1 (K rows 10,11 zero) + bias tile.
    for (int i = lane; i < 12 * 16; i += 32) {
      const int k = i >> 4, col = i & 15;
      const int gcol = t * 16 + col;
      wt[(((k >> 1) * 16 + col) << 1) + (k & 1)] =
          (k < 2 * RR && gcol < H1) ? w1[k * H1 + gcol] : 0.f;
    }
    if (lane < 16) bt[lane] = (t * 16 + lane < H1) ? b1[t * 16 + lane] : 0.f;
    __syncthreads();

    v8f acc;
    const float bv = bt[col0];
#pragma unroll
    for (int r = 0; r < 8; ++r) acc[r] = bv;
#pragma unroll
    for (int s = 0; s < 3; ++s) {
      const int ka = s * 4 + hi * 2;
      v2f a  = *(const v2f*)&fpad[m][ka];
      v2f bm = *(const v2f*)&wt[((ka >> 1) * 16 + col0) << 1];
      acc = __builtin_amdgcn_wmma_f32_16x16x4_f32(
          false, a, false, bm, (short)0, acc, false, false);
    }
    if (lane < 16) {
#pragma unroll
      for (int r = 0; r < 8; ++r) h1[r][t * 16 + lane] = acc[r];
    }
    __syncthreads();
  }

  // ---- Layer 2: [16 x 200] @ [200 x 100] -> h2[16 x 112] ----
  for (int t = 0; t < 7; ++t) {
    stage_btile(wt, w2, 200, H2, t, lane);
    if (lane < 16) bt[lane] = (t * 16 + lane < H2) ? b2[t * 16 + lane] : 0.f;
    __syncthreads();

    v8f acc;
    const float bv = bt[col0];
#pragma unroll
    for (int r = 0; r < 8; ++r) acc[r] = bv;
#pragma unroll 10
    for (int s = 0; s < 50; ++s) {
      const int ka = s * 4 + hi * 2;
      v2f a  = *(const v2f*)&h1[m][ka];
      v2f bm = *(const v2f*)&wt[((ka >> 1) * 16 + col0) << 1];
      acc = __builtin_amdgcn_wmma_f32_16x16x4_f32(
          false, a, false, bm, (short)0, acc, false, false);
    }
    if (lane < 16) {
#pragma unroll
      for (int r = 0; r < 8; ++r) h2[r][t * 16 + lane] = acc[r];
    }
    __syncthreads();
  }

  // ---- Layer 3: [16 x 100] @ [100 x 2] -> out[8 x 2] ----
  {
    stage_btile(wt, w3, 100, CC, 0, lane);
    if (lane < 16) bt[lane] = (lane < CC) ? b3[lane] : 0.f;
    __syncthreads();

    v8f acc;
    const float bv = bt[col0];
#pragma unroll
    for (int r = 0; r < 8; ++r) acc[r] = bv;
#pragma unroll 5
    for (int s = 0; s < 25; ++s) {
      const int ka = s * 4 + hi * 2;
      v2f a  = *(const v2f*)&h2[m][ka];
      v2f bm = *(const v2f*)&wt[((ka >> 1) * 16 + col0) << 1];
      acc = __builtin_amdgcn_wmma_f32_16x16x4_f32(
          false, a, false, bm, (short)0, acc, false, false);
    }
    if (lane < CC) {
#pragma unroll
      for (int r = 0; r < 8; ++r) out[r * CC + lane] = acc[r];  // [8,1,2]
    }
  }
}

// ---------------------------------------------------------------------------
// Launch
// ---------------------------------------------------------------------------
extern "C" void kernel_launch(void* const* d_in, const int* in_sizes, int n_in,
                              void* d_out, int out_size, void* d_ws, size_t ws_size,
                              hipStream_t stream) {
  const float* x      = (const float*)d_in[0];
  const float* conv_w = (const float*)d_in[1];
  const float* conv_b = (const float*)d_in[2];
  const float* w1     = (const float*)d_in[3];
  const float* b1     = (const float*)d_in[4];
  const float* w2     = (const float*)d_in[5];
  const float* b2     = (const float*)d_in[6];
  const float* w3     = (const float*)d_in[7];
  const float* b3     = (const float*)d_in[8];
  float* out = (float*)d_out;

  float* scores = (float*)d_ws;                  // B*N floats = 1.6 MB
  float* feat   = scores + (size_t)BB * NN;      // B*2R = 80 floats

  // Phase 1: streaming GEMV, one wave32 per instance, 8 waves/block.
  const int waves_per_blk = 256 / 32;
  const int nblk = (BB * NN + waves_per_blk - 1) / waves_per_blk;  // 50000
  chowder_scores<<<nblk, 256, 0, stream>>>(x, conv_w, conv_b, scores);

  // Phase 2: top/bottom-5 per batch row.
  chowder_topk<<<BB, 256, 0, stream>>>(scores, feat);

  // Phase 3: WMMA MLP, single wave.
  chowder_mlp<<<1, 32, 0, stream>>>(feat, w1, b1, w2, b2, w3, b3, out);
}